// VisionTransformer_17514876633982
// MI455X (gfx1250) — compile-verified
//
#include <hip/hip_runtime.h>
#include <hip/hip_bf16.h>

#define SEQ   2048
#define EMB   4
#define BATCH 16
#define FFD   16
#define NTOK  17
#define NLAYER 4

typedef __attribute__((ext_vector_type(2))) float v2f;
typedef __attribute__((ext_vector_type(8))) float v8f;

// ---------------------------------------------------------------------------
// Embedding gather: x[b,s,:] = emb[mask ? 16 : patch]
// ---------------------------------------------------------------------------
__global__ void __launch_bounds__(256) embed_kernel(
    const int* __restrict__ patches, const int* __restrict__ mask,
    const float* __restrict__ emb, float* __restrict__ x)
{
    int i = blockIdx.x * 256 + threadIdx.x;           // token index, 0..B*S-1
    if (i >= BATCH * SEQ) return;
    int row = mask[i] ? 16 : patches[i];
    float4 v = *(const float4*)&emb[row * EMB];
    *(float4*)&x[i * EMB] = v;
}

// ---------------------------------------------------------------------------
// One fused encoder layer:
//   attention (WMMA f32 16x16x4 for K·Qᵀ, single-pass exp-sum softmax in VALU)
//   + residual + LN1 + FFN(4->16->4, relu) + residual + LN2
// One workgroup = one batch b, 128 queries (8 waves x 16-query tiles).
// Entire x[b] (32KB) + mask-bias (8KB) + 63-entry rel bias live in LDS.
// The rel-pos bias index clip(q-k+31,0,62) saturates for |q-k|>=31, so all
// k-tiles outside [qbase-32, qbase+48) use a scalar bias constant (fast path,
// no per-element index math / LDS gathers). Loop bounds are made scalar via
// readfirstlane so the phase loops stay on the SALU branch pipe.
// ---------------------------------------------------------------------------
__global__ void __launch_bounds__(256) encoder_layer_kernel(
    const float* __restrict__ xin, float* __restrict__ xout,
    const int*   __restrict__ mask,
    const float* __restrict__ rel,                      // (63, 4) for this layer
    const float* __restrict__ w1, const float* __restrict__ b1,
    const float* __restrict__ w2, const float* __restrict__ b2,
    const float* __restrict__ ln1w, const float* __restrict__ ln1b,
    const float* __restrict__ ln2w, const float* __restrict__ ln2b)
{
    __shared__ __align__(16) float s_x[SEQ * EMB];      // 32 KB: x[b] (q = k = v)
    __shared__ float s_mb[SEQ];                         // 8 KB: 0 or -1e30 per key
    __shared__ float s_bias[64];                        // summed rel-pos bias

    const int b   = blockIdx.y;
    const int tid = threadIdx.x;

    // cooperative LDS fill
    const float* xb_ = xin + (size_t)b * SEQ * EMB;
    for (int i = tid; i < SEQ; i += 256) {
        *(float4*)&s_x[i * EMB] = *(const float4*)&xb_[i * EMB];
        s_mb[i] = mask[b * SEQ + i] ? 0.0f : -1e30f;
    }
    if (tid < 63) {
        const float* rp = rel + tid * EMB;
        s_bias[tid] = rp[0] + rp[1] + rp[2] + rp[3];
    }
    __syncthreads();

    const int wave  = tid >> 5;
    const int lane  = tid & 31;
    const int qbase = blockIdx.x * 128 + wave * 16;
    const int qg    = qbase + (lane & 15);
    const int ehalf = (lane & 16) ? 2 : 0;              // K-dim half owned by this lane
    const int khalf = (lane & 16) ? 8 : 0;              // M(=key)-row half in D layout

    // B operand: Qᵀ (4x16), pre-scaled by 1/sqrt(E) = 0.5
    v2f bq;
    {
        const float* qp = &s_x[qg * EMB + ehalf];
        bq[0] = qp[0] * 0.5f;
        bq[1] = qp[1] * 0.5f;
    }

    float den = 0.0f, num0 = 0.0f, num1 = 0.0f, num2 = 0.0f, num3 = 0.0f;

    // saturated bias constants for distant tiles
    const float bias_past = s_bias[62];   // q - k >= 31  -> index clamps to 62
    const float bias_fut  = s_bias[0];    // q - k <= -31 -> index clamps to 0

    // scalar (SGPR) phase boundaries, multiples of 16
    const int qbase_s   = __builtin_amdgcn_readfirstlane(qbase);
    int past_end  = qbase_s - 32; if (past_end  < 0)   past_end  = 0;
    int fut_start = qbase_s + 48; if (fut_start > SEQ) fut_start = SEQ;

    // fast tile: uniform saturated bias, no index math / bias gathers
    auto do_tile_fast = [&](int kbase, float cb) {
        v2f a;
        const float* kp = &s_x[(kbase + (lane & 15)) * EMB + ehalf];
        a[0] = kp[0];
        a[1] = kp[1];
        v8f st = {};
        st = __builtin_amdgcn_wmma_f32_16x16x4_f32(
                 false, a, false, bq, (short)0, st, false, false);
        #pragma unroll
        for (int r = 0; r < 8; ++r) {
            const int kg = kbase + khalf + r;
            float e = __expf(st[r] + cb + s_mb[kg]);
            den += e;
            const float* vp = &s_x[kg * EMB];
            num0 += e * vp[0];
            num1 += e * vp[1];
            num2 += e * vp[2];
            num3 += e * vp[3];
        }
    };

    for (int kbase = 0; kbase < past_end; kbase += 16)
        do_tile_fast(kbase, bias_past);

    for (int kbase = past_end; kbase < fut_start; kbase += 16) {
        v2f a;
        const float* kp = &s_x[(kbase + (lane & 15)) * EMB + ehalf];
        a[0] = kp[0];
        a[1] = kp[1];
        v8f st = {};
        st = __builtin_amdgcn_wmma_f32_16x16x4_f32(
                 false, a, false, bq, (short)0, st, false, false);
        #pragma unroll
        for (int r = 0; r < 8; ++r) {
            const int kg = kbase + khalf + r;
            int di = qg - kg + 31;
            di = di < 0 ? 0 : (di > 62 ? 62 : di);
            float e = __expf(st[r] + s_bias[di] + s_mb[kg]);
            den += e;
            const float* vp = &s_x[kg * EMB];
            num0 += e * vp[0];
            num1 += e * vp[1];
            num2 += e * vp[2];
            num3 += e * vp[3];
        }
    }

    for (int kbase = fut_start; kbase < SEQ; kbase += 16)
        do_tile_fast(kbase, bias_fut);

    // fold the two key halves (lane ^ 16 holds the other 8 rows of each tile)
    den  += __shfl_xor(den,  16, 32);
    num0 += __shfl_xor(num0, 16, 32);
    num1 += __shfl_xor(num1, 16, 32);
    num2 += __shfl_xor(num2, 16, 32);
    num3 += __shfl_xor(num3, 16, 32);

    if (lane < 16) {
        const float inv = 1.0f / den;
        const float* xp = &s_x[qg * EMB];
        float y[EMB];
        y[0] = xp[0] + num0 * inv;
        y[1] = xp[1] + num1 * inv;
        y[2] = xp[2] + num2 * inv;
        y[3] = xp[3] + num3 * inv;

        // LN1
        float mean = 0.25f * (y[0] + y[1] + y[2] + y[3]);
        float var = 0.0f;
        #pragma unroll
        for (int d = 0; d < EMB; ++d) { float t = y[d] - mean; var += t * t; }
        float rstd = __frsqrt_rn(var * 0.25f + 1e-5f);
        float yn[EMB];
        #pragma unroll
        for (int d = 0; d < EMB; ++d)
            yn[d] = (y[d] - mean) * rstd * ln1w[d] + ln1b[d];

        // FFN 4 -> 16 -> 4 with relu
        float z[EMB] = { b2[0], b2[1], b2[2], b2[3] };
        #pragma unroll
        for (int j = 0; j < FFD; ++j) {
            float h = b1[j];
            #pragma unroll
            for (int d = 0; d < EMB; ++d) h += yn[d] * w1[d * FFD + j];
            h = fmaxf(h, 0.0f);
            #pragma unroll
            for (int d = 0; d < EMB; ++d) z[d] += h * w2[j * EMB + d];
        }

        // residual + LN2
        float y2[EMB];
        #pragma unroll
        for (int d = 0; d < EMB; ++d) y2[d] = yn[d] + z[d];
        mean = 0.25f * (y2[0] + y2[1] + y2[2] + y2[3]);
        var = 0.0f;
        #pragma unroll
        for (int d = 0; d < EMB; ++d) { float t = y2[d] - mean; var += t * t; }
        rstd = __frsqrt_rn(var * 0.25f + 1e-5f);
        float4 o;
        o.x = (y2[0] - mean) * rstd * ln2w[0] + ln2b[0];
        o.y = (y2[1] - mean) * rstd * ln2w[1] + ln2b[1];
        o.z = (y2[2] - mean) * rstd * ln2w[2] + ln2b[2];
        o.w = (y2[3] - mean) * rstd * ln2w[3] + ln2b[3];
        *(float4*)&xout[((size_t)b * SEQ + qg) * EMB] = o;
    }
}

// ---------------------------------------------------------------------------
// Final classifier: logits[b,s,t] = x[b,s,:] . fc_w[t,:] + fc_b[t]
// ---------------------------------------------------------------------------
__global__ void __launch_bounds__(256) logits_kernel(
    const float* __restrict__ x, const float* __restrict__ fcw,
    const float* __restrict__ fcb, float* __restrict__ out)
{
    int i = blockIdx.x * 256 + threadIdx.x;            // token index
    if (i >= BATCH * SEQ) return;
    float4 xv = *(const float4*)&x[i * EMB];
    #pragma unroll
    for (int t = 0; t < NTOK; ++t) {
        out[(size_t)i * NTOK + t] =
            fcb[t] + xv.x * fcw[t * EMB + 0] + xv.y * fcw[t * EMB + 1]
                   + xv.z * fcw[t * EMB + 2] + xv.w * fcw[t * EMB + 3];
    }
}

// ---------------------------------------------------------------------------
extern "C" void kernel_launch(void* const* d_in, const int* in_sizes, int n_in,
                              void* d_out, int out_size, void* d_ws, size_t ws_size,
                              hipStream_t stream) {
    const int*   patches = (const int*)  d_in[0];
    const int*   mask    = (const int*)  d_in[1];
    const float* emb     = (const float*)d_in[2];
    const float* rel     = (const float*)d_in[3];   // (L, 63, 4)
    const float* w1      = (const float*)d_in[4];   // (L, 4, 16)
    const float* b1      = (const float*)d_in[5];   // (L, 16)
    const float* w2      = (const float*)d_in[6];   // (L, 16, 4)
    const float* b2      = (const float*)d_in[7];   // (L, 4)
    const float* ln1w    = (const float*)d_in[8];
    const float* ln1b    = (const float*)d_in[9];
    const float* ln2w    = (const float*)d_in[10];
    const float* ln2b    = (const float*)d_in[11];
    const float* fcw     = (const float*)d_in[12];  // (17, 4)
    const float* fcb     = (const float*)d_in[13];  // (17,)
    float* out = (float*)d_out;

    float* xa = (float*)d_ws;                       // (B, S, E)
    float* xb = xa + (size_t)BATCH * SEQ * EMB;     // ping-pong buffer

    const int ntok_grid = (BATCH * SEQ) / 256;

    embed_kernel<<<ntok_grid, 256, 0, stream>>>(patches, mask, emb, xa);

    float* cur = xa;
    float* nxt = xb;
    for (int l = 0; l < NLAYER; ++l) {
        encoder_layer_kernel<<<dim3(SEQ / 128, BATCH), 256, 0, stream>>>(
            cur, nxt, mask,
            rel  + (size_t)l * 63 * EMB,
            w1   + (size_t)l * EMB * FFD, b1 + (size_t)l * FFD,
            w2   + (size_t)l * FFD * EMB, b2 + (size_t)l * EMB,
            ln1w + (size_t)l * EMB, ln1b + (size_t)l * EMB,
            ln2w + (size_t)l * EMB, ln2b + (size_t)l * EMB);
        float* t = cur; cur = nxt; nxt = t;
    }

    logits_kernel<<<ntok_grid, 256, 0, stream>>>(cur, fcw, fcb, out);
}